// Net_45741401702526
// MI455X (gfx1250) — compile-verified
//
#include <hip/hip_runtime.h>
#include <math.h>

// Problem constants (match reference)
#define NB     256                 // graphs (B)
#define NPER   512                 // nodes per graph
#define NTOT   (NB * NPER)         // 131072 nodes
#define DEG_E  16
#define EPER   (NPER * DEG_E)      // 8192 edges per graph
#define ETOT   (NB * EPER)         // 2097152 edges
#define HDIM   128
#define KSEL   256                 // kept nodes per graph (top-k)
#define NCLS   10

typedef float v2f __attribute__((ext_vector_type(2)));
typedef float v8f __attribute__((ext_vector_type(8)));
typedef int   v4i __attribute__((ext_vector_type(4)));

typedef __attribute__((address_space(1))) v4i* gas_b128_t;  // global b128 chunk
typedef __attribute__((address_space(3))) v4i* las_b128_t;  // LDS b128 chunk

#if defined(__HIP_DEVICE_COMPILE__) && __has_builtin(__builtin_amdgcn_global_load_async_to_lds_b128)
#define HAVE_ASYNC_LDS 1
#else
#define HAVE_ASYNC_LDS 0
#endif

// ---------------------------------------------------------------------------
// degree / normalization
// ---------------------------------------------------------------------------
__global__ void k_init_deg(float* deg) {
    int i = blockIdx.x * blockDim.x + threadIdx.x;
    if (i < NTOT) deg[i] = 1.0f;   // self loop
}
__global__ void k_count_deg(const int* __restrict__ cols, float* deg) {
    int e = blockIdx.x * blockDim.x + threadIdx.x;
    if (e < ETOT) atomicAdd(&deg[cols[e]], 1.0f);
}
__global__ void k_finish_dinv(float* deg) {
    int i = blockIdx.x * blockDim.x + threadIdx.x;
    if (i < NTOT) deg[i] = rsqrtf(deg[i]);
}

// ---------------------------------------------------------------------------
// WMMA fp32 GEMM:  C[M x 128] = A[M x K] @ W[K x 128]  (+bias, +relu optional)
// One wave computes one 16x16 tile via V_WMMA_F32_16X16X4_F32.
// Block = 256 threads = 8 waves covering 16 rows x 128 cols. grid.x = M/16.
// The 16xK A-tile (<=32KB) is staged in LDS once per block via the gfx1250
// async global->LDS copy (ASYNCcnt) so the 8 waves don't re-read it 8x from
// global (the ss matrix is 256MB > 192MB L2, so amplification would hit HBM).
//
// Fragment layouts (CDNA5 ISA 7.12.2, wave32):
//   A 16x4 f32 : lanes 0-15 hold M=lane {K=k,k+1}; lanes 16-31 hold {K=k+2,k+3}
//   B 4x16 f32 : vgpr0: lanes0-15 K=k row, lanes16-31 K=k+1; vgpr1: K=k+2/k+3
//   C 16x16 f32: vgpr r: lanes0-15 M=r, lanes16-31 M=r+8; N=lane&15
// ---------------------------------------------------------------------------
template <int K>
__global__ void k_gemm_wmma(const float* __restrict__ A,
                            const float* __restrict__ W,
                            const float* __restrict__ bias,
                            float* __restrict__ C,
                            int relu) {
    __shared__ float As[16 * K];               // 8KB (K=128) / 32KB (K=512)
    const int tid   = threadIdx.x;
    const int mbase = blockIdx.x * 16;

    // rows mbase..mbase+15 are one contiguous 16*K block of A
    const float* Abase = A + (size_t)mbase * K;
    constexpr int NCHUNK = 16 * K / 4;         // float4 chunks
#if HAVE_ASYNC_LDS
    for (int c = tid; c < NCHUNK; c += 256) {
        __builtin_amdgcn_global_load_async_to_lds_b128(
            (gas_b128_t)(Abase + 4 * c),
            (las_b128_t)(As + 4 * c),
            0, 0);
    }
    __builtin_amdgcn_s_wait_asynccnt(0);
#else
    for (int c = tid; c < NCHUNK; c += 256) {
        float4 v = *(const float4*)(Abase + 4 * c);
        *(float4*)(As + 4 * c) = v;
    }
#endif
    __syncthreads();

    const int wave  = tid >> 5;
    const int lane  = tid & 31;
    const int h     = lane >> 4;               // lane half 0/1
    const int l     = lane & 15;
    const int nbase = wave * 16;

    const float* Arow = As + l * K;            // LDS
    v8f c = {};
#pragma unroll 8
    for (int k = 0; k < K; k += 4) {
        v2f a, b;
        a.x = Arow[k + 2 * h];
        a.y = Arow[k + 2 * h + 1];
        b.x = W[(size_t)(k + h)     * HDIM + nbase + l];
        b.y = W[(size_t)(k + 2 + h) * HDIM + nbase + l];
        c = __builtin_amdgcn_wmma_f32_16x16x4_f32(false, a, false, b,
                                                  (short)0, c, false, false);
    }
#pragma unroll
    for (int r = 0; r < 8; ++r) {
        int row = mbase + r + 8 * h;
        int col = nbase + l;
        float v = c[r];
        if (bias) v += bias[col];
        if (relu) v = fmaxf(v, 0.0f);
        C[(size_t)row * HDIM + col] = v;
    }
}

// ---------------------------------------------------------------------------
// Per-graph edge aggregation in LDS (320KB WGP LDS holds the 256KB tile).
// out[col] = relu( sum_edges h[row]*dinv[row]*dinv[col] + h[col]*dinv[col]^2 + b )
// One workgroup per graph; 256 threads; ds_add_f32 LDS atomics for scatter.
// Lane l owns features 4l..4l+3 -> one float4 gather + 4 LDS atomics per edge.
// ---------------------------------------------------------------------------
__global__ void k_aggregate(const float* __restrict__ hlin,
                            const int* __restrict__ rows,
                            const int* __restrict__ cols,
                            const float* __restrict__ dinv,
                            const float* __restrict__ bias,
                            float* __restrict__ out) {
    extern __shared__ float acc[];            // NPER * HDIM floats = 256 KB
    const int g   = blockIdx.x;
    const int tid = threadIdx.x;              // 256 threads, 8 waves

    // self-loop initialization: acc[i][f] = h[i][f] * dinv[i]^2  (float4)
    for (int idx4 = tid; idx4 < NPER * HDIM / 4; idx4 += 256) {
        int i = idx4 >> 5;                    // 32 float4 per node
        int node = g * NPER + i;
        float di = dinv[node];
        float4 v = ((const float4*)hlin)[(size_t)node * 32 + (idx4 & 31)];
        v.x *= di * di; v.y *= di * di; v.z *= di * di; v.w *= di * di;
        ((float4*)acc)[idx4] = v;
    }
    __syncthreads();

    const int wave = tid >> 5, lane = tid & 31;
    const int e0 = g * EPER;
    for (int e = wave; e < EPER; e += 8) {
        int r    = rows[e0 + e];
        int ccol = cols[e0 + e];
        float norm = dinv[r] * dinv[ccol];
        if (e + 8 < EPER)  // emits global_prefetch_b8 for the next gather row
            __builtin_prefetch(hlin + (size_t)rows[e0 + e + 8] * HDIM + 4 * lane, 0, 0);
        float4 v = ((const float4*)hlin)[(size_t)r * 32 + lane];
        int lc = ((ccol & (NPER - 1)) << 7) + 4 * lane;
        atomicAdd(&acc[lc + 0], v.x * norm);  // ds_add_f32
        atomicAdd(&acc[lc + 1], v.y * norm);
        atomicAdd(&acc[lc + 2], v.z * norm);
        atomicAdd(&acc[lc + 3], v.w * norm);
    }
    __syncthreads();

    for (int idx = tid; idx < NPER * HDIM; idx += 256) {
        int i = idx >> 7, f = idx & 127;
        float v = acc[idx] + bias[f];
        out[(size_t)(g * NPER + i) * HDIM + f] = fmaxf(v, 0.0f);
    }
}

// ---------------------------------------------------------------------------
// slin[i] = dot(h[i][:], Wp[:,0])   (one wave per node)
// ---------------------------------------------------------------------------
__global__ void k_dotvec(const float* __restrict__ H,
                         const float* __restrict__ Wp,
                         float* __restrict__ out) {
    int wave = threadIdx.x >> 5, lane = threadIdx.x & 31;
    int i = blockIdx.x * 8 + wave;
    if (i >= NTOT) return;
    const float* row = H + (size_t)i * HDIM;
    float s = 0.0f;
#pragma unroll
    for (int f = lane; f < HDIM; f += 32) s += row[f] * Wp[f];
#pragma unroll
    for (int off = 16; off > 0; off >>= 1) s += __shfl_down(s, off, 32);
    if (lane == 0) out[i] = s;
}

// score = edge-aggregated 1-dim GCNConv of slin (self-loop + bias in init)
__global__ void k_score_init(const float* __restrict__ slin,
                             const float* __restrict__ dinv,
                             const float* __restrict__ bp,
                             float* __restrict__ score) {
    int i = blockIdx.x * blockDim.x + threadIdx.x;
    if (i < NTOT) {
        float di = dinv[i];
        score[i] = slin[i] * di * di + bp[0];
    }
}
__global__ void k_score_edges(const int* __restrict__ rows,
                              const int* __restrict__ cols,
                              const float* __restrict__ slin,
                              const float* __restrict__ dinv,
                              float* __restrict__ score) {
    int e = blockIdx.x * blockDim.x + threadIdx.x;
    if (e < ETOT) {
        int r = rows[e], c = cols[e];
        atomicAdd(&score[c], slin[r] * dinv[r] * dinv[c]);
    }
}

// ---------------------------------------------------------------------------
// Per-graph top-K (bitonic sort of 512 scores in LDS) + tanh-weighted
// max/mean readout. max/mean are order-invariant so the sorted set suffices.
// Writes z[g][zoff .. zoff+255] = [max(128) || mean(128)].
// ---------------------------------------------------------------------------
__global__ void k_topk_readout(const float* __restrict__ score,
                               const float* __restrict__ hfeat,
                               float* __restrict__ z, int zoff) {
    __shared__ float sv[NPER];
    __shared__ int   si[NPER];
    const int g = blockIdx.x, tid = threadIdx.x;   // 256 threads

    for (int i = tid; i < NPER; i += 256) { sv[i] = score[g * NPER + i]; si[i] = i; }
    __syncthreads();

    // bitonic sort, descending
    for (int k = 2; k <= NPER; k <<= 1) {
        for (int j = k >> 1; j > 0; j >>= 1) {
            for (int i = tid; i < NPER; i += 256) {
                int ixj = i ^ j;
                if (ixj > i) {
                    bool desc = ((i & k) == 0);
                    bool sw = desc ? (sv[i] < sv[ixj]) : (sv[i] > sv[ixj]);
                    if (sw) {
                        float tv = sv[i]; sv[i] = sv[ixj]; sv[ixj] = tv;
                        int   ti = si[i]; si[i] = si[ixj]; si[ixj] = ti;
                    }
                }
            }
            __syncthreads();
        }
    }

    // readout over top KSEL nodes: 128 threads, one per feature
    if (tid < HDIM) {
        float m = -3.402823466e+38f, sum = 0.0f;
        for (int k = 0; k < KSEL; ++k) {
            int node = si[k];
            float w = tanhf(sv[k]);
            float v = hfeat[(size_t)(g * NPER + node) * HDIM + tid] * w;
            m = fmaxf(m, v);
            sum += v;
        }
        float* zr = z + (size_t)g * 512 + zoff;
        zr[tid]        = m;
        zr[HDIM + tid] = sum * (1.0f / (float)KSEL);
    }
}

// ---------------------------------------------------------------------------
// head: out[row][c] = log_softmax( z2[row] @ Wl3 + bl3 )  (one wave per row)
// ---------------------------------------------------------------------------
__global__ void k_head(const float* __restrict__ z2,
                       const float* __restrict__ Wl3,
                       const float* __restrict__ bl3,
                       float* __restrict__ out) {
    __shared__ float lg[NCLS];
    const int row = blockIdx.x, lane = threadIdx.x;   // block = 32
    if (lane < NCLS) {
        float s = bl3[lane];
        const float* zr = z2 + (size_t)row * HDIM;
        for (int f = 0; f < HDIM; ++f) s += zr[f] * Wl3[f * NCLS + lane];
        lg[lane] = s;
    }
    __syncthreads();
    if (lane < NCLS) {
        float m = lg[0];
#pragma unroll
        for (int i = 1; i < NCLS; ++i) m = fmaxf(m, lg[i]);
        float sum = 0.0f;
#pragma unroll
        for (int i = 0; i < NCLS; ++i) sum += expf(lg[i] - m);
        out[(size_t)row * NCLS + lane] = lg[lane] - m - logf(sum);
    }
}

// ---------------------------------------------------------------------------
// launcher
// ---------------------------------------------------------------------------
extern "C" void kernel_launch(void* const* d_in, const int* in_sizes, int n_in,
                              void* d_out, int out_size, void* d_ws, size_t ws_size,
                              hipStream_t stream) {
    const float* x   = (const float*)d_in[0];
    const float* ss  = (const float*)d_in[1];
    const int*   e1  = (const int*)d_in[2];
    const int*   e2  = (const int*)d_in[3];
    const float* W1  = (const float*)d_in[4];
    const float* b1  = (const float*)d_in[5];
    const float* Wp1 = (const float*)d_in[6];
    const float* bp1 = (const float*)d_in[7];
    const float* W2  = (const float*)d_in[8];
    const float* b2  = (const float*)d_in[9];
    const float* Wp2 = (const float*)d_in[10];
    const float* bp2 = (const float*)d_in[11];
    const float* Wl1 = (const float*)d_in[12];
    const float* bl1 = (const float*)d_in[13];
    const float* Wl3 = (const float*)d_in[14];
    const float* bl3 = (const float*)d_in[15];
    float* out = (float*)d_out;

    // workspace layout (floats)
    float* buf_lin = (float*)d_ws;                       // NTOT*128
    float* buf_h   = buf_lin + (size_t)NTOT * HDIM;      // NTOT*128
    float* dinv    = buf_h   + (size_t)NTOT * HDIM;      // NTOT
    float* slin    = dinv  + NTOT;                       // NTOT
    float* score   = slin  + NTOT;                       // NTOT
    float* zbuf    = score + NTOT;                       // NB*512
    float* z2      = zbuf  + (size_t)NB * 512;           // NB*128

    // 256 KB dynamic LDS for the per-graph accumulator (WGP LDS = 320 KB)
    const int aggLds = NPER * HDIM * (int)sizeof(float);
    (void)hipFuncSetAttribute((const void*)k_aggregate,
                              hipFuncAttributeMaxDynamicSharedMemorySize, aggLds);

    auto run_branch = [&](const float* feat, int K, const float* W,
                          const float* bias, const float* Wp, const float* bp,
                          const int* eidx, int zoff) {
        const int* rows = eidx;          // edge_index[0]
        const int* cols = eidx + ETOT;   // edge_index[1]
        k_init_deg   <<<NTOT / 256, 256, 0, stream>>>(dinv);
        k_count_deg  <<<ETOT / 256, 256, 0, stream>>>(cols, dinv);
        k_finish_dinv<<<NTOT / 256, 256, 0, stream>>>(dinv);
        if (K == 128)
            k_gemm_wmma<128><<<NTOT / 16, 256, 0, stream>>>(feat, W, nullptr,
                                                            buf_lin, 0);
        else
            k_gemm_wmma<512><<<NTOT / 16, 256, 0, stream>>>(feat, W, nullptr,
                                                            buf_lin, 0);
        k_aggregate  <<<NB, 256, aggLds, stream>>>(buf_lin, rows, cols, dinv,
                                                   bias, buf_h);
        k_dotvec     <<<NTOT / 8, 256, 0, stream>>>(buf_h, Wp, slin);
        k_score_init <<<NTOT / 256, 256, 0, stream>>>(slin, dinv, bp, score);
        k_score_edges<<<ETOT / 256, 256, 0, stream>>>(rows, cols, slin, dinv, score);
        k_topk_readout<<<NB, 256, 0, stream>>>(score, buf_h, zbuf, zoff);
    };

    run_branch(x,  HDIM, W1, b1, Wp1, bp1, e1, 0);
    run_branch(ss, NPER, W2, b2, Wp2, bp2, e2, 256);

    // z [256,512] @ Wl1 [512,128] + bl1, relu  (WMMA)
    k_gemm_wmma<512><<<NB / 16, 256, 0, stream>>>(zbuf, Wl1, bl1, z2, 1);
    // head + log_softmax
    k_head<<<NB, 32, 0, stream>>>(z2, Wl3, bl3, out);
}